// YoloLoss_37538014167663
// MI455X (gfx1250) — compile-verified
//
#include <hip/hip_runtime.h>
#include <hip/hip_bf16.h>

// Problem constants (match reference)
#define BATCH 256
#define NCLS  1019
#define GRID  7
#define GG    49            // GRID*GRID
#define NOBJ  16
#define NCH   1024          // 5 + NCLS
#define SLAB  (NCH * GG)    // 50176 f32 elements per batch (200704 bytes)
#define SMALL 1024          // bytes reserved for small shared arrays before slab

typedef float v2f  __attribute__((ext_vector_type(2)));
typedef float v8f  __attribute__((ext_vector_type(8)));
typedef unsigned int u32x4 __attribute__((ext_vector_type(4)));
typedef int i32x8 __attribute__((ext_vector_type(8)));
typedef int i32x4 __attribute__((ext_vector_type(4)));

__global__ __launch_bounds__(256)
void yolo_loss_main(const float* __restrict__ outputs,
                    const float* __restrict__ boxes,
                    const int*   __restrict__ labels,
                    const int*   __restrict__ coords,
                    const float* __restrict__ objm,
                    float*       __restrict__ partial)
{
    extern __shared__ unsigned char smem[];
    int*   s_pos  = (int*)(smem);              // [16]
    float* s_max  = (float*)(smem + 64);       // [16]
    float* s_part = (float*)(smem + 128);      // [8][16] per-wave WMMA partial sums
    float* s_ce   = (float*)(smem + 640);      // [16]
    float* s_bb   = (float*)(smem + 704);      // [64]
    float* s_obj  = (float*)(smem + 960);      // [1]
    float* slab   = (float*)(smem + SMALL);    // [SLAB] staged batch slab

    const int b    = blockIdx.x;
    const int tid  = threadIdx.x;
    const int wave = tid >> 5;
    const int lane = tid & 31;

    // ---- TDM: DMA this batch's full [1024 x 49] f32 slab into LDS ----------
    // One descriptor, 1D tile of SLAB contiguous f32 elements. Issued by wave 0
    // only (TDM ignores EXEC; the branch is wave-uniform so other waves skip).
    if (tid < 32) {
        unsigned long long ga = (unsigned long long)(const void*)outputs
                              + (unsigned long long)b * (unsigned long long)(SLAB * 4);
        unsigned lds_addr = __builtin_amdgcn_groupstaticsize() + SMALL;
        u32x4 g0;
        g0[0] = 1u;                                     // count = 1 (valid), user mode
        g0[1] = lds_addr;                               // LDS byte address of tile
        g0[2] = (unsigned)(ga & 0xFFFFFFFFull);         // global_addr[31:0]
        g0[3] = (unsigned)((ga >> 32) & 0x1FFFFFFull)   // global_addr[56:32]
              | 0x80000000u;                            // type = 2 ("image")
        i32x8 g1;
        g1[0] = 0x00020000;                             // data_size = 2 (4 bytes)
        g1[1] = (int)((SLAB & 0xFFFF) << 16);           // tensor_dim0[15:0]
        g1[2] = (int)(((SLAB >> 16) & 0xFFFF) | (1 << 16)); // dim0[31:16] | tensor_dim1=1
        g1[3] = (int)((SLAB & 0xFFFF) << 16);           // tile_dim0 = SLAB
        g1[4] = 1;                                      // tile_dim1 = 1, tile_dim2 = 0
        g1[5] = SLAB;                                   // tensor_dim0_stride[31:0]
        g1[6] = (int)((SLAB & 0xFFFF) << 16);           // stride0[47:32]=0 | stride1[15:0]
        g1[7] = 0;                                      // stride1[47:16]
        i32x4 g2 = {0, 0, 0, 0};                        // group2: unused (<=2D tensor)
        i32x4 g3 = {0, 0, 0, 0};                        // group3: unused (<=2D tensor)
        i32x8 gx = {0, 0, 0, 0, 0, 0, 0, 0};            // extra group (clang-23 form): null
        __builtin_amdgcn_tensor_load_to_lds(g0, g1, g2, g3, gx, 0);
    }
    // Object cell positions while DMA is in flight.
    if (tid < NOBJ) {
        int x = coords[(b * NOBJ + tid) * 2 + 0];
        int y = coords[(b * NOBJ + tid) * 2 + 1];
        s_pos[tid] = y * GRID + x;
    }
    __builtin_amdgcn_s_wait_tensorcnt(0);   // issuing wave waits; others pass
    __syncthreads();                        // publish LDS to all waves

    // ---- Pass 1: per-object max over 1019 class logits (wave w -> obj 2w,2w+1)
    // LDS stride between lanes is 49 banks, gcd(49,64)=1 -> conflict-free.
    for (int t = 0; t < 2; ++t) {
        const int obj = wave * 2 + t;
        const int p   = s_pos[obj];
        float m = -3.402823466e38f;
        for (int c = lane; c < NCLS; c += 32)
            m = fmaxf(m, slab[(5 + c) * GG + p]);
        for (int off = 16; off > 0; off >>= 1)
            m = fmaxf(m, __shfl_xor(m, off, 32));
        if (lane == 0) s_max[obj] = m;
    }
    __syncthreads();

    // ---- Pass 2: WMMA f32 16x16x4 with all-ones B => per-object sum of exp.
    // A layout (32-bit A, 16x4): lanes 0-15 hold (K0,K1), lanes 16-31 (K2,K3),
    // row M = lane & 15. B = ones so D[m][n] = sum_k A[m][k] for every n.
    {
        const int   obj  = lane & 15;
        const int   p    = s_pos[obj];
        const float mx   = s_max[obj];
        const int   kofs = (lane < 16) ? 0 : 2;
        v8f d = {0.f, 0.f, 0.f, 0.f, 0.f, 0.f, 0.f, 0.f};
        v2f onesB; onesB.x = 1.0f; onesB.y = 1.0f;
        for (int q = wave; q < 255; q += 8) {     // 255 chunks of K=4 cover 1019 (pad=0)
            const int c0 = q * 4 + kofs;
            const int c1 = c0 + 1;
            v2f a;
            a.x = (c0 < NCLS) ? __expf(slab[(5 + c0) * GG + p] - mx) : 0.0f;
            a.y = (c1 < NCLS) ? __expf(slab[(5 + c1) * GG + p] - mx) : 0.0f;
            d = __builtin_amdgcn_wmma_f32_16x16x4_f32(false, a, false, onesB,
                                                      (short)0, d, false, false);
        }
        // D layout: VGPR j, lanes 0-15 -> row j, lanes 16-31 -> row j+8. Column
        // 0 suffices (all columns equal). Fixed LDS slots -> deterministic.
        if (lane == 0) {
#pragma unroll
            for (int m2 = 0; m2 < 8; ++m2) s_part[wave * 16 + m2] = d[m2];
        } else if (lane == 16) {
#pragma unroll
            for (int m2 = 0; m2 < 8; ++m2) s_part[wave * 16 + 8 + m2] = d[m2];
        }
    }
    __syncthreads();

    // ---- Finalize: three independent wave-regions run concurrently ---------
    if (tid < 16) {
        // Class CE per object: log(sum_exp) + max - logit[label]
        float s = 0.0f;
#pragma unroll
        for (int w = 0; w < 8; ++w) s += s_part[w * 16 + tid];  // fixed order
        const int   lab = labels[b * NOBJ + tid] - 1;
        const float xl  = slab[(5 + lab) * GG + s_pos[tid]];
        s_ce[tid] = __logf(s) + s_max[tid] - xl;
    } else if (tid >= 32 && tid < 64) {
        // Objectness soft-CE over the 49-cell channel-0 map (wave 1).
        const int   i  = tid - 32;
        const float x0 = slab[i];
        const bool  hi = (i + 32) < GG;
        const float x1 = hi ? slab[i + 32] : -3.402823466e38f;
        float m = fmaxf(x0, x1);
        for (int off = 16; off > 0; off >>= 1) m = fmaxf(m, __shfl_xor(m, off, 32));
        float e = __expf(x0 - m) + (hi ? __expf(x1 - m) : 0.0f);
        for (int off = 16; off > 0; off >>= 1) e += __shfl_xor(e, off, 32);
        const float ls = __logf(e);
        float ts = objm[b * GG + i] * (x0 - m - ls);
        if (hi) ts += objm[b * GG + i + 32] * (x1 - m - ls);
        for (int off = 16; off > 0; off >>= 1) ts += __shfl_xor(ts, off, 32);
        if (i == 0) s_obj[0] = -ts;
    } else if (tid >= 64 && tid < 128) {
        // Box L2 loss: 16 objects x 4 coords (waves 2-3).
        const int   k    = tid - 64;
        const int   n    = k >> 2, j = k & 3;
        const float pred = slab[(1 + j) * GG + s_pos[n]];
        // jnp.round = round-half-even -> rintf (RNE); match op order of reference.
        const float tgt  = rintf(boxes[(b * NOBJ + n) * 4 + j] / 448.0f * 10.0f) / 10.0f;
        const float dd   = tgt - pred;
        s_bb[k] = dd * dd;
    }
    __syncthreads();

    if (tid == 0) {
        float tot = s_obj[0];
        for (int i = 0; i < 64; ++i)  tot += s_bb[i];   // fixed order
        for (int n = 0; n < NOBJ; ++n) tot += s_ce[n];  // fixed order
        partial[b] = tot;   // all three terms get the common /256 at the end
    }
}

__global__ __launch_bounds__(256)
void yolo_loss_reduce(const float* __restrict__ partial, float* __restrict__ out)
{
    __shared__ float s_red[8];
    const int tid = threadIdx.x;
    float p = partial[tid];
    for (int off = 16; off > 0; off >>= 1) p += __shfl_xor(p, off, 32);
    if ((tid & 31) == 0) s_red[tid >> 5] = p;
    __syncthreads();
    if (tid == 0) {
        float t = 0.0f;
        for (int i = 0; i < 8; ++i) t += s_red[i];      // fixed order
        out[0] = t * (1.0f / 256.0f);                   // exact /B (power of two)
    }
}

extern "C" void kernel_launch(void* const* d_in, const int* in_sizes, int n_in,
                              void* d_out, int out_size, void* d_ws, size_t ws_size,
                              hipStream_t stream) {
    (void)in_sizes; (void)n_in; (void)out_size; (void)ws_size;
    const float* outputs = (const float*)d_in[0];   // [256,1024,7,7] f32
    const float* boxes   = (const float*)d_in[1];   // [256,16,4] f32
    const int*   labels  = (const int*)d_in[2];     // [256,16] i32
    const int*   coords  = (const int*)d_in[3];     // [256,16,2] i32
    const float* objm    = (const float*)d_in[4];   // [256,49] f32
    float* out  = (float*)d_out;
    float* part = (float*)d_ws;                     // 256 floats of scratch

    const size_t shmem = (size_t)SMALL + (size_t)SLAB * sizeof(float); // ~201 KB < 320 KB/WGP
    yolo_loss_main<<<dim3(BATCH), dim3(256), shmem, stream>>>(
        outputs, boxes, labels, coords, objm, part);
    yolo_loss_reduce<<<dim3(1), dim3(256), 0, stream>>>(part, out);
}